// SphericalSelfWarp_67224828117125
// MI455X (gfx1250) — compile-verified
//
#include <hip/hip_runtime.h>
#include <hip/hip_bf16.h>

// ---------------------------------------------------------------------------
// SphericalSelfWarp for MI455X (gfx1250, wave32, WMMA)
//   Kernel 1: fp32 WMMA GEMMs (h = relu(W1*u), value = Wv*u) + VALU flow head
//   Kernel 2: spherical warp (periodic lon, polar flip, border lat clamp)
// ---------------------------------------------------------------------------

#define HH    361
#define WW    720
#define HWPX  (HH * WW)          // 259920 = 16 * 16245 (exact)
#define CIN   96
#define COUT  96
#define HEADS 4
#define CI    (COUT / HEADS)     // 24
#define NTILE (HWPX / 16)        // 16245 pixel tiles of width 16
#define WPB   8                  // waves per block (256 threads)
#define KSTEPS (CIN / 4)         // 24 WMMA k-steps
#define MT    (COUT / 16)        // 6 M-tiles

typedef __attribute__((ext_vector_type(2))) float v2f;
typedef __attribute__((ext_vector_type(8))) float v8f;

#define PI_F     3.14159265358979323846f
#define TWOPI_F  6.28318530717958647692f

// ---------------------------------------------------------------------------
// Kernel 1: per-wave 16-pixel tile.  D = A(16x4) x B(4x16) + C via
// v_wmma_f32_16x16x4_f32.  Fragment layouts per CDNA5 ISA 7.12.2:
//   A 16x4 : lanes 0-15 -> M=lane,   VGPR0=K+0, VGPR1=K+1
//            lanes16-31 -> M=lane-16,VGPR0=K+2, VGPR1=K+3
//   B 4x16 : lanes 0-15 -> N=lane,   VGPR0=K+0, VGPR1=K+1
//            lanes16-31 -> N=lane-16,VGPR0=K+2, VGPR1=K+3
//   C 16x16: VGPR r -> M=r (lanes 0-15) / M=r+8 (lanes 16-31), N=lane%16
// ---------------------------------------------------------------------------
__global__ __launch_bounds__(256, 1)
void flow_value_gemm_kernel(const float* __restrict__ u,
                            const float* __restrict__ w1,
                            const float* __restrict__ b1,
                            const float* __restrict__ w2,
                            const float* __restrict__ b2,
                            const float* __restrict__ wv,
                            const float* __restrict__ bv,
                            float* __restrict__ flow,
                            float* __restrict__ value)
{
    __shared__ float lds_h[WPB * CIN * 16];   // 8 waves * 96ch * 16px = 48 KB

    const int wave = threadIdx.x >> 5;        // wave32
    const int lane = threadIdx.x & 31;
    const int half = lane >> 4;               // 0: lanes 0-15, 1: lanes 16-31
    const int l16  = lane & 15;

    const int tile = blockIdx.x * WPB + wave; // wave-uniform
    if (tile >= NTILE) return;                // whole-wave exit (EXEC stays full)
    const int pix0 = tile * 16;

    v8f acc_h[MT];
    v8f acc_v[MT];
#pragma unroll
    for (int t = 0; t < MT; ++t) { acc_h[t] = (v8f){}; acc_v[t] = (v8f){}; }

    const float* __restrict__ up = u + pix0 + l16;   // B column base for this lane

#pragma unroll 4
    for (int ks = 0; ks < KSTEPS; ++ks) {
        const int k = ks * 4 + half * 2;             // this lane's K pair

        if (ks + 2 < KSTEPS)                          // pull next u lines toward WGP$
            __builtin_prefetch(up + (k + 8) * HWPX, 0, 0);

        v2f bfrag;
        bfrag.x = up[(k + 0) * HWPX];
        bfrag.y = up[(k + 1) * HWPX];

#pragma unroll
        for (int t = 0; t < MT; ++t) {
            const int m = t * 16 + l16;              // A row for this lane
            v2f a1 = *(const v2f*)(w1 + m * CIN + k);
            acc_h[t] = __builtin_amdgcn_wmma_f32_16x16x4_f32(
                false, a1, false, bfrag, (short)0, acc_h[t], false, false);
            v2f av = *(const v2f*)(wv + m * CIN + k);
            acc_v[t] = __builtin_amdgcn_wmma_f32_16x16x4_f32(
                false, av, false, bfrag, (short)0, acc_v[t], false, false);
        }
    }

    // Epilogue: bias + relu -> h into LDS; bias -> value into workspace.
    float* __restrict__ hsh = &lds_h[wave * CIN * 16];
#pragma unroll
    for (int t = 0; t < MT; ++t) {
#pragma unroll
        for (int r = 0; r < 8; ++r) {
            const int m = t * 16 + r + half * 8;     // C/D row for this half-wave
            float hv = acc_h[t][r] + b1[m];
            hv = fmaxf(hv, 0.0f);
            hsh[m * 16 + l16] = hv;
            value[m * HWPX + pix0 + l16] = acc_v[t][r] + bv[m];
        }
    }

    // Flow head: M=8, K=96 — tiny (4% of FLOPs), do as VALU dot products.
    // Lane handles pixel l16 and 4 flow dirs (half selects dirs 0-3 / 4-7).
    const int dbase = half * 4;
    float s0 = 0.f, s1 = 0.f, s2 = 0.f, s3 = 0.f;
#pragma unroll 8
    for (int k = 0; k < CIN; ++k) {
        const float hv = hsh[k * 16 + l16];          // LDS broadcast read
        s0 = fmaf(hv, w2[(dbase + 0) * CIN + k], s0);
        s1 = fmaf(hv, w2[(dbase + 1) * CIN + k], s1);
        s2 = fmaf(hv, w2[(dbase + 2) * CIN + k], s2);
        s3 = fmaf(hv, w2[(dbase + 3) * CIN + k], s3);
    }
    flow[(dbase + 0) * HWPX + pix0 + l16] = s0 + b2[dbase + 0];
    flow[(dbase + 1) * HWPX + pix0 + l16] = s1 + b2[dbase + 1];
    flow[(dbase + 2) * HWPX + pix0 + l16] = s2 + b2[dbase + 2];
    flow[(dbase + 3) * HWPX + pix0 + l16] = s3 + b2[dbase + 3];
}

// ---------------------------------------------------------------------------
// Kernel 2: spherical warp + bilinear sample.
// The reference's padded-width algebra collapses: ix = px (Wp-1 == W) and the
// padded column W equals column 0.  gy handled with polar flip + border clamp.
// value[] (100 MB) fits in the 192 MB L2, so the 4-corner gathers stay on-chip.
// ---------------------------------------------------------------------------
__global__ __launch_bounds__(256, 1)
void warp_sample_kernel(const float* __restrict__ flow,
                        const float* __restrict__ value,
                        const float* __restrict__ base_grid,
                        const float* __restrict__ lat_top_p,
                        const float* __restrict__ lat_range_p,
                        float* __restrict__ out)
{
    const int idx = blockIdx.x * 256 + threadIdx.x;
    if (idx >= HEADS * HWPX) return;
    const int hd  = idx / HWPX;
    const int rem = idx - hd * HWPX;                 // y*W + x

    float lon = base_grid[rem * 2 + 0] + flow[(2 * hd + 0) * HWPX + rem];
    float lat = base_grid[rem * 2 + 1] + flow[(2 * hd + 1) * HWPX + rem];

    // lon mod 2*pi (python %: result in [0, 2*pi))
    lon = lon - floorf(lon * (1.0f / TWOPI_F)) * TWOPI_F;
    float gx = lon * (1.0f / PI_F) - 1.0f;

    const float lat_top   = lat_top_p[0];
    const float lat_range = lat_range_p[0];
    float gy = 2.0f * (lat_top - lat) / lat_range - 1.0f;

    const bool pn = gy > 1.0f;
    const bool ps = gy < -1.0f;
    if (pn) gy = 2.0f - gy;
    if (ps) gy = -2.0f - gy;
    if (pn || ps) gx += 1.0f;                        // 180-degree lon shift

    // px = ((gx+1)*(W-1)/2) mod W  ; ix == px after padded-width round trip
    float px = (gx + 1.0f) * 0.5f * (float)(WW - 1);
    px = px - floorf(px * (1.0f / (float)WW)) * (float)WW;
    if (px >= (float)WW) px -= (float)WW;            // roundoff guard
    if (px < 0.0f) px = 0.0f;

    float iy = (gy + 1.0f) * 0.5f * (float)(HH - 1); // border clamp in y
    iy = fminf(fmaxf(iy, 0.0f), (float)(HH - 1));

    const float x0f = floorf(px);
    const float y0f = floorf(iy);
    const float wx  = px - x0f;
    const float wy  = iy - y0f;

    int x0 = (int)x0f;                               // in [0, W-1]
    int y0 = (int)y0f;
    int x1 = x0 + 1; if (x1 >= WW) x1 = 0;           // padded col W == col 0
    int y1 = y0 + 1; if (y1 > HH - 1) y1 = HH - 1;   // border

    const float w00 = (1.0f - wx) * (1.0f - wy);
    const float w01 = wx * (1.0f - wy);
    const float w10 = (1.0f - wx) * wy;
    const float w11 = wx * wy;

    const int i00 = y0 * WW + x0;
    const int i01 = y0 * WW + x1;
    const int i10 = y1 * WW + x0;
    const int i11 = y1 * WW + x1;

    const float* __restrict__ vbase = value + hd * CI * HWPX;
    float* __restrict__       obase = out   + hd * CI * HWPX;

#pragma unroll 4
    for (int c = 0; c < CI; ++c) {
        const float* __restrict__ vp = vbase + c * HWPX;
        const float r = vp[i00] * w00 + vp[i01] * w01 +
                        vp[i10] * w10 + vp[i11] * w11;
        obase[c * HWPX + rem] = r;
    }
}

// ---------------------------------------------------------------------------
extern "C" void kernel_launch(void* const* d_in, const int* in_sizes, int n_in,
                              void* d_out, int out_size, void* d_ws, size_t ws_size,
                              hipStream_t stream)
{
    (void)in_sizes; (void)n_in; (void)out_size; (void)ws_size;

    const float* u   = (const float*)d_in[0];   // [1,96,361,720]
    const float* w1  = (const float*)d_in[1];   // [96,96]
    const float* b1  = (const float*)d_in[2];   // [96]
    const float* w2  = (const float*)d_in[3];   // [8,96]
    const float* b2  = (const float*)d_in[4];   // [8]
    const float* wv  = (const float*)d_in[5];   // [96,96]
    const float* bv  = (const float*)d_in[6];   // [96]
    const float* bg  = (const float*)d_in[7];   // [361,720,2]
    const float* lt  = (const float*)d_in[8];   // scalar
    const float* lr  = (const float*)d_in[9];   // scalar

    float* flow  = (float*)d_ws;                // [8, HW]
    float* value = flow + (size_t)(2 * HEADS) * HWPX;   // [96, HW]
    float* out   = (float*)d_out;               // [1, 96, 361, 720]

    const int grid1 = (NTILE + WPB - 1) / WPB;                 // 2031 blocks
    flow_value_gemm_kernel<<<grid1, 256, 0, stream>>>(
        u, w1, b1, w2, b2, wv, bv, flow, value);

    const int grid2 = (HEADS * HWPX + 255) / 256;              // 4062 blocks
    warp_sample_kernel<<<grid2, 256, 0, stream>>>(
        flow, value, bg, lt, lr, out);
}